// layer_class_mean_29480655520074
// MI455X (gfx1250) — compile-verified
//
#include <hip/hip_runtime.h>
#include <stdint.h>

// ---------------------------------------------------------------------------
// Segment mean (per-class mean of rows): out[c, :] = mean over rows with
// classes[i]==c.  Bandwidth-bound (1.02 GB read -> ~45us floor @ 23.3 TB/s).
// Strategy: D split into 8 slices of 16 cols; per-block LDS-privatized
// accumulator (1024x16 f32) fed by CDNA5 TDM (tensor_load_to_lds) async DMA,
// double buffered, ds_add_f32 atomics into LDS, global_atomic_add_f32 flush.
// ---------------------------------------------------------------------------

typedef uint32_t u32;
typedef uint64_t u64;
typedef __attribute__((ext_vector_type(4))) u32 u32x4;
typedef __attribute__((ext_vector_type(8))) int i32x8;
typedef __attribute__((ext_vector_type(4))) int i32x4;

#define C_CLASSES 1024
#define D_COLS    128
#define SLICE_W   16          // columns per block (64 bytes per row segment)
#define NSLICES   (D_COLS / SLICE_W)
#define BLOCKS_PER_SLICE 256
#define TILE_ROWS 256         // rows staged per TDM tile
#define THREADS   256         // 8 waves (wave32)

// LDS byte offset of a __shared__ pointer: flat LDS pointers carry the
// wave-relative LDS offset in their low 32 bits.
__device__ __forceinline__ u32 lds_off(const void* p) {
  return (u32)(uintptr_t)p;
}

// Issue one TDM 2D tile load: tile_d0 x tile_d1 elements (data_size code 2 ->
// 4 bytes/elem), row stride stride0 elements, into LDS at lds_byte_off.
// D# packing per CDNA5 ISA ch.8 (group0 128b, group1 256b; groups 2/3 zero ->
// <=2D tensor, iterate/gather/pad disabled, no multicast, no atomic barrier).
// Toolchain here is clang-23 (therock-10.0): 6-arg builtin
//   (u32x4 g0, i32x8 g1, i32x4 g2, i32x4 g3, i32x8 ext, i32 cpol)
__device__ __forceinline__ void tdm_load_2d(u32 lds_byte_off, u64 gaddr,
                                            u32 tile_d0, u32 tile_d1,
                                            u32 tens_d0, u32 tens_d1,
                                            u64 stride0_elems) {
  u32x4 g0;
  g0[0] = 1u;                                    // count=1 (valid user D#)
  g0[1] = lds_byte_off;                          // lds_addr [63:32]
  g0[2] = (u32)gaddr;                            // global_addr[31:0]
  g0[3] = (u32)((gaddr >> 32) & 0x01FFFFFFu)     // global_addr[56:32]
        | (2u << 30);                            // type=2 ("image")
  i32x8 g1;
  g1[0] = (int)(2u << 16);                       // data_size=2 (4B), mask=0
  g1[1] = (int)((tens_d0 & 0xFFFFu) << 16);      // tensor_dim0[15:0]
  g1[2] = (int)(((tens_d0 >> 16) & 0xFFFFu)      // tensor_dim0[31:16]
        | ((tens_d1 & 0xFFFFu) << 16));          // tensor_dim1[15:0]
  g1[3] = (int)(((tens_d1 >> 16) & 0xFFFFu)      // tensor_dim1[31:16]
        | ((tile_d0 & 0xFFFFu) << 16));          // tile_dim0
  g1[4] = (int)(tile_d1 & 0xFFFFu);              // tile_dim1 (tile_dim2=0)
  g1[5] = (int)(u32)(stride0_elems & 0xFFFFFFFFu);  // tensor_dim0_stride lo
  g1[6] = (int)(u32)((stride0_elems >> 32) & 0xFFFFu); // stride0 hi, stride1=0
  g1[7] = 0;
  i32x4 z4 = {0, 0, 0, 0};                       // groups 2/3: zero (<=2D)
  i32x8 z8 = {0, 0, 0, 0, 0, 0, 0, 0};           // extended group: unused
  __builtin_amdgcn_tensor_load_to_lds(g0, g1, z4, z4, z8, 0);
}

__global__ void __launch_bounds__(THREADS)
zero_kernel(float* __restrict__ out, float* __restrict__ counts) {
  int i = blockIdx.x * THREADS + threadIdx.x;
  if (i < C_CLASSES * D_COLS) out[i] = 0.0f;
  if (i < C_CLASSES) counts[i] = 0.0f;
}

__global__ void __launch_bounds__(THREADS)
accum_kernel(const float* __restrict__ x, const int* __restrict__ classes,
             float* __restrict__ sums, float* __restrict__ counts,
             int N, int rows_per_block) {
  __shared__ float s_acc[C_CLASSES * SLICE_W];          // 64 KB
  __shared__ float s_cnt[C_CLASSES];                    //  4 KB
  __shared__ float s_x[2][TILE_ROWS * SLICE_W];         // 32 KB (staging)
  __shared__ int   s_cls[2][TILE_ROWS];                 //  2 KB (staging)

  const int slice = (int)blockIdx.x / BLOCKS_PER_SLICE;
  const int rb    = (int)blockIdx.x % BLOCKS_PER_SLICE;
  const int row0  = rb * rows_per_block;
  const int rowE  = min(N, row0 + rows_per_block);
  const int nrows = max(0, rowE - row0);
  const int ntiles = (nrows + TILE_ROWS - 1) / TILE_ROWS;

  for (int i = threadIdx.x; i < C_CLASSES * SLICE_W; i += THREADS) s_acc[i] = 0.0f;
  for (int i = threadIdx.x; i < C_CLASSES; i += THREADS) s_cnt[i] = 0.0f;

  const int wid  = threadIdx.x >> 5;   // wave id (wave32)
  const int lane = threadIdx.x & 31;
  const int col  = lane & 15;
  const int sub  = lane >> 4;          // 2 rows handled per wave iteration

  auto issue = [&](int t, int buf) {
    const int r0 = row0 + t * TILE_ROWS;
    const int rc = min(TILE_ROWS, rowE - r0);
    // x tile: rc rows x 16 cols of f32, row stride 128 elements
    u64 gx = (u64)(uintptr_t)x + ((u64)r0 * D_COLS + (u64)slice * SLICE_W) * 4u;
    tdm_load_2d(lds_off(&s_x[buf][0]), gx,
                /*tile*/ SLICE_W, (u32)rc, /*tensor*/ SLICE_W, (u32)rc,
                /*stride0*/ (u64)D_COLS);
    // classes tile: rc contiguous i32 (1D: tile_dim1 = 0 -> unused)
    u64 gc = (u64)(uintptr_t)classes + (u64)r0 * 4u;
    tdm_load_2d(lds_off(&s_cls[buf][0]), gc,
                /*tile*/ (u32)rc, 0u, /*tensor*/ (u32)rc, 1u,
                /*stride0*/ (u64)rc);
  };

  // Prologue: prefetch tiles 0 and 1 (2 tensor ops each; in-order completion).
  if (wid == 0) {
    if (ntiles > 0) issue(0, 0);
    if (ntiles > 1) issue(1, 1);
  }

  for (int t = 0; t < ntiles; ++t) {
    if (wid == 0) {
      if (t + 1 < ntiles) __builtin_amdgcn_s_wait_tensorcnt(2); // next still in flight
      else                __builtin_amdgcn_s_wait_tensorcnt(0); // drain last tile
    }
    __syncthreads();  // publish staged buffer t&1 to all 8 waves

    const int r0 = row0 + t * TILE_ROWS;
    const int rc = min(TILE_ROWS, rowE - r0);
    const float* xs = &s_x[t & 1][0];
    const int*   cs = &s_cls[t & 1][0];
    for (int r = wid * 2 + sub; r < rc; r += 16) {
      const int c = cs[r];
      atomicAdd(&s_acc[c * SLICE_W + col], xs[r * SLICE_W + col]); // ds_add_f32
      if (col == 0 && slice == 0) atomicAdd(&s_cnt[c], 1.0f);
    }
    __syncthreads();  // buffer t&1 fully consumed, safe to overwrite

    if (wid == 0 && t + 2 < ntiles) issue(t + 2, t & 1);
  }
  __syncthreads();

  // Flush privatized accumulator to global sums (d_out) with f32 atomics.
  for (int i = threadIdx.x; i < C_CLASSES * SLICE_W; i += THREADS) {
    const int c = i >> 4, cc = i & 15;
    unsafeAtomicAdd(&sums[c * D_COLS + slice * SLICE_W + cc], s_acc[i]);
  }
  if (slice == 0) {
    for (int i = threadIdx.x; i < C_CLASSES; i += THREADS)
      unsafeAtomicAdd(&counts[i], s_cnt[i]);
  }
}

__global__ void __launch_bounds__(THREADS)
div_kernel(float* __restrict__ out, const float* __restrict__ counts) {
  int i = blockIdx.x * THREADS + threadIdx.x;
  if (i < C_CLASSES * D_COLS) out[i] = out[i] / counts[i >> 7]; // i/128 -> class
}

extern "C" void kernel_launch(void* const* d_in, const int* in_sizes, int n_in,
                              void* d_out, int out_size, void* d_ws, size_t ws_size,
                              hipStream_t stream) {
  const float* x   = (const float*)d_in[0];
  const int*   cls = (const int*)d_in[1];
  // d_in[2] is total_classes (==1024, baked into C_CLASSES)
  float* out    = (float*)d_out;       // 1024*128 f32: used as the sums buffer
  float* counts = (float*)d_ws;        // 1024 f32 scratch

  const int N = in_sizes[0] / D_COLS;
  const int rows_per_block = (N + BLOCKS_PER_SLICE - 1) / BLOCKS_PER_SLICE;

  zero_kernel<<<(C_CLASSES * D_COLS + THREADS - 1) / THREADS, THREADS, 0, stream>>>(out, counts);
  accum_kernel<<<NSLICES * BLOCKS_PER_SLICE, THREADS, 0, stream>>>(x, cls, out, counts,
                                                                   N, rows_per_block);
  div_kernel<<<(C_CLASSES * D_COLS + THREADS - 1) / THREADS, THREADS, 0, stream>>>(out, counts);
}